// AttentionGNN_77068893160012
// MI455X (gfx1250) — compile-verified
//
#include <hip/hip_runtime.h>
#include <math.h>

// ---------------- WMMA types ----------------
typedef float v2f __attribute__((ext_vector_type(2)));
typedef float v8f __attribute__((ext_vector_type(8)));

#define IN_DIM  8
#define HID_DIM 64
#define OUT_DIM 112

// ============================================================
// GEMM: out{q,k,v,s} = h @ W{q,k,v,s} + b{..}
// One wave (32 lanes) per (16-row tile, weight matrix).
// Block = 256 threads = 8 waves = 2 tiles x 4 matrices.
// V_WMMA_F32_16X16X4_F32 (exact fp32 path). Compile-time CIN/COUT
// so A fragments stay in VGPRs across all N-tiles and the K-chain
// fully unrolls.
// ============================================================
template <int CIN, int COUT>
__global__ void __launch_bounds__(256)
k_gemm_qkvs(const float* __restrict__ h, int ntiles,
            const float* __restrict__ Wq, const float* __restrict__ bq,
            const float* __restrict__ Wk, const float* __restrict__ bk,
            const float* __restrict__ Wv, const float* __restrict__ bv,
            const float* __restrict__ Ws, const float* __restrict__ bs,
            float* __restrict__ oq, float* __restrict__ ok,
            float* __restrict__ ov, float* __restrict__ os)
{
    constexpr int NK = CIN / 4;                // K-steps
    constexpr int NT = COUT / 16;              // N-tiles

    const int wave = threadIdx.x >> 5;         // 0..7
    const int lane = threadIdx.x & 31;
    const int mat  = wave & 3;                 // which weight matrix
    const int tile = blockIdx.x * 2 + (wave >> 2);
    if (tile >= ntiles) return;                // uniform per wave -> EXEC all-1s

    const float* W; const float* b; float* out;
    if      (mat == 0) { W = Wq; b = bq; out = oq; }
    else if (mat == 1) { W = Wk; b = bk; out = ok; }
    else if (mat == 2) { W = Wv; b = bv; out = ov; }
    else               { W = Ws; b = bs; out = os; }

    const int half  = lane >> 4;               // 0 -> K=0,1 ; 1 -> K=2,3
    const int khalf = half * 2;
    const int row   = tile * 16 + (lane & 15); // A-matrix row for this lane
    const float* hrow = h + (size_t)row * CIN;
    const int col0  = lane & 15;               // B/C/D column for this lane
    const int rbase = tile * 16 + half * 8;    // C/D row base for this lane

    // Hoist the whole A tile into registers once (reused for all N-tiles).
    v2f a[NK];
    #pragma unroll
    for (int t = 0; t < NK; ++t) {
        a[t].x = hrow[4 * t + khalf];
        a[t].y = hrow[4 * t + khalf + 1];
    }

    #pragma unroll
    for (int nt = 0; nt < NT; ++nt) {
        const int col = nt * 16 + col0;
        const float bias = b[col];
        v8f c;
        #pragma unroll
        for (int i = 0; i < 8; ++i) c[i] = bias;   // bias replicated per column

        #pragma unroll
        for (int t = 0; t < NK; ++t) {
            v2f bb;
            bb.x = W[(4 * t + khalf)     * COUT + col];
            bb.y = W[(4 * t + khalf + 1) * COUT + col];
            c = __builtin_amdgcn_wmma_f32_16x16x4_f32(
                    false, a[t], false, bb, (short)0, c, false, false);
        }
        #pragma unroll
        for (int i = 0; i < 8; ++i)
            out[(size_t)(rbase + i) * COUT + col] = c[i];
    }
}

// ---------------- monotone float<->uint for atomic max ----------------
__device__ __forceinline__ unsigned f2u_mono(float f) {
    unsigned u = __float_as_uint(f);
    return (u & 0x80000000u) ? ~u : (u | 0x80000000u);
}
__device__ __forceinline__ float u2f_mono(unsigned u) {
    return (u & 0x80000000u) ? __uint_as_float(u & 0x7FFFFFFFu)
                             : __uint_as_float(~u);
}

__global__ void k_init_nodes(unsigned* __restrict__ mbits,
                             float* __restrict__ denom, int N)
{
    int i = blockIdx.x * blockDim.x + threadIdx.x;
    if (i < N) { mbits[i] = 0u; denom[i] = 0.0f; }
}

// scores[i] = <q[dst], k[src]> * scale ; segment max into mbits[dst]
__global__ void k_scores(const int* __restrict__ src, const int* __restrict__ dst,
                         const float* __restrict__ q, const float* __restrict__ kk,
                         float* __restrict__ e, unsigned* __restrict__ mbits,
                         int E, int C, float scale)
{
    int i = blockIdx.x * blockDim.x + threadIdx.x;
    if (i >= E) return;
    const int d = dst[i], s = src[i];
    const float4* qd = reinterpret_cast<const float4*>(q + (size_t)d * C);
    const float4* ks = reinterpret_cast<const float4*>(kk + (size_t)s * C);
    float acc = 0.0f;
    for (int c = 0; c < (C >> 2); ++c) {
        float4 a = qd[c], b = ks[c];
        acc += a.x * b.x + a.y * b.y + a.z * b.z + a.w * b.w;
    }
    acc *= scale;
    e[i] = acc;
    atomicMax(mbits + d, f2u_mono(acc));
}

// e[i] = exp(score - m[dst]) ; denom[dst] += e[i]
__global__ void k_expsum(const int* __restrict__ dst,
                         float* __restrict__ e,
                         const unsigned* __restrict__ mbits,
                         float* __restrict__ denom, int E)
{
    int i = blockIdx.x * blockDim.x + threadIdx.x;
    if (i >= E) return;
    const int d = dst[i];
    const float m = u2f_mono(mbits[d]);
    const float ex = __expf(e[i] - m);
    e[i] = ex;
    atomicAdd(denom + d, ex);
}

// out[dst] += (e[i]/denom[dst]) * v[src]   (float4 chunks, f32 atomics)
__global__ void k_aggregate(const int* __restrict__ src, const int* __restrict__ dst,
                            const float* __restrict__ e, const float* __restrict__ denom,
                            const float* __restrict__ v, float* __restrict__ out,
                            int E, int C)
{
    const int tpe = C >> 2;                    // float4 chunks per edge
    int tid = blockIdx.x * blockDim.x + threadIdx.x;
    int i = tid / tpe;
    if (i >= E) return;
    const int c4 = (tid - i * tpe) << 2;
    const int d = dst[i], s = src[i];
    const float alpha = e[i] / denom[d];
    const float4 vv = *reinterpret_cast<const float4*>(v + (size_t)s * C + c4);
    float* o = out + (size_t)d * C + c4;
    atomicAdd(o + 0, alpha * vv.x);
    atomicAdd(o + 1, alpha * vv.y);
    atomicAdd(o + 2, alpha * vv.z);
    atomicAdd(o + 3, alpha * vv.w);
}

__global__ void k_relu(float* __restrict__ h, int n)
{
    int i = blockIdx.x * blockDim.x + threadIdx.x;
    if (i < n) h[i] = fmaxf(h[i], 0.0f);
}

// ============================================================
// Host-side orchestration
// ============================================================
static inline int cdiv(long long a, long long b) { return (int)((a + b - 1) / b); }

extern "C" void kernel_launch(void* const* d_in, const int* in_sizes, int n_in,
                              void* d_out, int out_size, void* d_ws, size_t ws_size,
                              hipStream_t stream)
{
    const float* x  = (const float*)d_in[0];
    const int*   ei = (const int*)d_in[1];      // JAX default: int32
    const int N = in_sizes[0] / IN_DIM;
    const int E = in_sizes[1] / 2;
    const int* src = ei;
    const int* dst = ei + E;

    // weights: d_in[2 + 8*layer + {0..7}] = Wq,bq,Wk,bk,Wv,bv,Ws,bs
    const float* Wp[3][8];
    for (int l = 0; l < 3; ++l)
        for (int j = 0; j < 8; ++j)
            Wp[l][j] = (const float*)d_in[2 + 8 * l + j];

    // workspace layout (floats)
    const size_t NB = (size_t)N * OUT_DIM;      // max-width node buffer
    float* q     = (float*)d_ws;
    float* k     = q  + NB;
    float* v     = k  + NB;
    float* hA    = v  + NB;
    float* hB    = hA + NB;
    float* e     = hB + NB;                     // E floats
    float* denom = e  + E;                      // N floats
    unsigned* mbits = (unsigned*)(denom + N);   // N uints

    const int Cout[3] = { HID_DIM, HID_DIM, OUT_DIM };
    const float* hin[3] = { x, hA, hB };
    float* hout[3]      = { hA, hB, (float*)d_out };

    const int ntiles = N / 16;                  // 100000/16 = 6250
    const int gemmBlocks = cdiv(ntiles, 2);

    for (int l = 0; l < 3; ++l) {
        const int C = Cout[l];
        const float scale = 1.0f / sqrtf((float)C);

        k_init_nodes<<<cdiv(N, 256), 256, 0, stream>>>(mbits, denom, N);

        if (l == 0)
            k_gemm_qkvs<IN_DIM, HID_DIM><<<gemmBlocks, 256, 0, stream>>>(
                hin[l], ntiles,
                Wp[l][0], Wp[l][1], Wp[l][2], Wp[l][3],
                Wp[l][4], Wp[l][5], Wp[l][6], Wp[l][7],
                q, k, v, hout[l]);
        else if (l == 1)
            k_gemm_qkvs<HID_DIM, HID_DIM><<<gemmBlocks, 256, 0, stream>>>(
                hin[l], ntiles,
                Wp[l][0], Wp[l][1], Wp[l][2], Wp[l][3],
                Wp[l][4], Wp[l][5], Wp[l][6], Wp[l][7],
                q, k, v, hout[l]);
        else
            k_gemm_qkvs<HID_DIM, OUT_DIM><<<gemmBlocks, 256, 0, stream>>>(
                hin[l], ntiles,
                Wp[l][0], Wp[l][1], Wp[l][2], Wp[l][3],
                Wp[l][4], Wp[l][5], Wp[l][6], Wp[l][7],
                q, k, v, hout[l]);

        k_scores<<<cdiv(E, 256), 256, 0, stream>>>(src, dst, q, k, e, mbits, E, C, scale);

        k_expsum<<<cdiv(E, 256), 256, 0, stream>>>(dst, e, mbits, denom, E);

        k_aggregate<<<cdiv((long long)E * (C / 4), 256), 256, 0, stream>>>(
            src, dst, e, denom, v, hout[l], E, C);

        if (l < 2)
            k_relu<<<cdiv((long long)N * C, 256), 256, 0, stream>>>(hout[l], N * C);
    }
}